// GATLayer_32298154066115
// MI455X (gfx1250) — compile-verified
//
#include <hip/hip_runtime.h>

// GAT layer for MI455X (gfx1250, wave32).
// Memory-bound problem (~1.3 GB moved vs ~3.3 GFLOP), so GEMMs use the exact
// f32 WMMA path (v_wmma_f32_16x16x4_f32) rather than a lossy low-precision one.

#define N_NODES 50000
#define N_EDGES 800000
#define DIM     128     // IN_DIM == OUT_DIM == H*D
#define NH      4
#define HD      32

typedef __attribute__((ext_vector_type(2))) float v2f;
typedef __attribute__((ext_vector_type(8))) float v8f;

// Monotonic float<->uint mapping so atomicMax(uint) == float max.
__device__ __forceinline__ unsigned encode_f(float f) {
  unsigned u = __float_as_uint(f);
  return (u & 0x80000000u) ? ~u : (u | 0x80000000u);
}
__device__ __forceinline__ float decode_f(unsigned u) {
  unsigned b = (u & 0x80000000u) ? (u ^ 0x80000000u) : ~u;
  return __uint_as_float(b);
}

// ---------------------------------------------------------------------------
// Zero the segment-sum accumulator and softmax stats (harness poisons d_ws).
__global__ void gat_init_kernel(float* __restrict__ outacc,
                                unsigned* __restrict__ hmax,
                                float* __restrict__ hsum, int n) {
  int i = blockIdx.x * blockDim.x + threadIdx.x;
  int stride = gridDim.x * blockDim.x;
  for (int j = i; j < n; j += stride) outacc[j] = 0.0f;
  if (i < NH) { hmax[i] = 0u; hsum[i] = 0.0f; }
}

// ---------------------------------------------------------------------------
// out[m, o] = act( sum_k A[m,k] * W[o,k] + bias[o] ), M=50000, N=K=128.
// One 16x16 output tile per wave; 8 waves/block cover all 128 output cols.
// A-matrix 16x4 f32 layout: a[j] = A[r][k0 + j + 2*half]   (ISA 7.12.2)
// B-matrix 4x16 f32 layout: b[j] = B[k0 + j + 2*half][r] = W[o0+r][k0+j+2*half]
// C/D 16x16 f32 layout:     c[v] = D[v + 8*half][r]
__global__ void gat_gemm_kernel(const float* __restrict__ A,
                                const float* __restrict__ W,
                                const float* __restrict__ bias,
                                float* __restrict__ out,
                                int apply_elu) {
  const int lane = threadIdx.x & 31;
  const int wave = threadIdx.x >> 5;       // 0..7 -> output column tile
  const int r    = lane & 15;
  const int half = lane >> 4;
  const int m0   = blockIdx.x * 16;        // 3125 blocks * 16 = 50000 exactly
  const int o0   = wave * 16;

  const float bv = bias[o0 + r];
  v8f c;
#pragma unroll
  for (int v = 0; v < 8; ++v) c[v] = bv;   // fold bias into the accumulator

  const float* arow = A + (size_t)(m0 + r) * DIM + 2 * half;
  const float* wrow = W + (size_t)(o0 + r) * DIM + 2 * half;

#pragma unroll
  for (int k0 = 0; k0 < DIM; k0 += 4) {
    v2f av = *(const v2f*)(arow + k0);     // 8B aligned (k0 + 2*half is even)
    v2f bvv = *(const v2f*)(wrow + k0);
    c = __builtin_amdgcn_wmma_f32_16x16x4_f32(
        /*neg_a=*/false, av, /*neg_b=*/false, bvv,
        /*c_mod=*/(short)0, c, /*reuse_a=*/false, /*reuse_b=*/false);
  }

#pragma unroll
  for (int v = 0; v < 8; ++v) {
    float val = c[v];
    if (apply_elu) val = (val > 0.0f) ? val : (__expf(val) - 1.0f);
    out[(size_t)(m0 + v + 8 * half) * DIM + o0 + r] = val;
  }
}

// ---------------------------------------------------------------------------
// One wave per edge: lane holds feature d=lane of each head.
// alpha_logit[e,h] = <h_src[h,:], att[h,0:32]> + <h_dst[h,:], att[h,32:64]>
// Also accumulates the global per-head max (LDS-staged -> global atomicMax).
__global__ void gat_logits_kernel(const long long* __restrict__ ei,
                                  const float* __restrict__ xproj,
                                  const float* __restrict__ att,
                                  float* __restrict__ alpha,
                                  unsigned* __restrict__ hmax) {
  __shared__ unsigned smax[NH];
  if (threadIdx.x < NH) smax[threadIdx.x] = 0u;
  __syncthreads();

  const int lane = threadIdx.x & 31;
  const int e = (int)((blockIdx.x * blockDim.x + threadIdx.x) >> 5); // < E by grid
  const long long row = ei[e];
  const long long col = ei[N_EDGES + e];
  const float* hs = xproj + (size_t)row * DIM;
  const float* hd = xproj + (size_t)col * DIM;

  float lg[NH];
#pragma unroll
  for (int h = 0; h < NH; ++h) {
    float p = hs[h * HD + lane] * att[h * 2 * HD + lane]
            + hd[h * HD + lane] * att[h * 2 * HD + HD + lane];
#pragma unroll
    for (int off = 16; off > 0; off >>= 1) p += __shfl_xor(p, off, 32);
    lg[h] = p;                              // all 32 lanes hold the full sum
  }
  if (lane < NH) {
    alpha[(size_t)e * NH + lane] = lg[lane];
    atomicMax(&smax[lane], encode_f(lg[lane]));
  }
  __syncthreads();
  if (threadIdx.x < NH) atomicMax(&hmax[threadIdx.x], smax[threadIdx.x]);
}

// ---------------------------------------------------------------------------
// alpha <- exp(alpha - max_h); accumulate per-head global sums.
__global__ void gat_expsum_kernel(float* __restrict__ alpha,
                                  const unsigned* __restrict__ hmax,
                                  float* __restrict__ hsum) {
  __shared__ float ssum[NH];
  if (threadIdx.x < NH) ssum[threadIdx.x] = 0.0f;
  __syncthreads();

  float m[NH], acc[NH];
#pragma unroll
  for (int h = 0; h < NH; ++h) { m[h] = decode_f(hmax[h]); acc[h] = 0.0f; }

  const int stride = gridDim.x * blockDim.x;
  for (int e = blockIdx.x * blockDim.x + threadIdx.x; e < N_EDGES; e += stride) {
#pragma unroll
    for (int h = 0; h < NH; ++h) {
      float v = __expf(alpha[(size_t)e * NH + h] - m[h]);
      alpha[(size_t)e * NH + h] = v;
      acc[h] += v;
    }
  }
#pragma unroll
  for (int h = 0; h < NH; ++h) atomicAdd(&ssum[h], acc[h]);
  __syncthreads();
  if (threadIdx.x < NH) atomicAdd(&hsum[threadIdx.x], ssum[threadIdx.x]);
}

// ---------------------------------------------------------------------------
// Normalize alpha in place and scatter alpha * h_src into outacc[col].
__global__ void gat_scatter_kernel(const long long* __restrict__ ei,
                                   const float* __restrict__ xproj,
                                   float* __restrict__ alpha,
                                   const float* __restrict__ hsum,
                                   float* __restrict__ outacc) {
  const int lane = threadIdx.x & 31;
  const int e = (int)((blockIdx.x * blockDim.x + threadIdx.x) >> 5);
  const long long row = ei[e];
  const long long col = ei[N_EDGES + e];

  float a[NH];
#pragma unroll
  for (int h = 0; h < NH; ++h)
    a[h] = alpha[(size_t)e * NH + h] / hsum[h];  // reads precede the write below
  if (lane < NH) alpha[(size_t)e * NH + lane] = a[lane];

  const float* hs = xproj + (size_t)row * DIM;
  float* dst = outacc + (size_t)col * DIM;
#pragma unroll
  for (int h = 0; h < NH; ++h)
    atomicAdd(&dst[h * HD + lane], a[h] * hs[h * HD + lane]);
}

// ---------------------------------------------------------------------------
extern "C" void kernel_launch(void* const* d_in, const int* in_sizes, int n_in,
                              void* d_out, int out_size, void* d_ws, size_t ws_size,
                              hipStream_t stream) {
  (void)in_sizes; (void)n_in; (void)out_size; (void)ws_size;

  const float*     x   = (const float*)d_in[0];
  const long long* ei  = (const long long*)d_in[1];   // int64 edge_index (2,E)
  const float*     Ww  = (const float*)d_in[2];
  const float*     Wb  = (const float*)d_in[3];
  const float*     att = (const float*)d_in[4];
  const float*     Ow  = (const float*)d_in[5];
  const float*     Ob  = (const float*)d_in[6];

  float* out   = (float*)d_out;                         // N*128 floats
  float* alpha = out + (size_t)N_NODES * DIM;           // E*4 floats (in d_out)

  float*    xproj  = (float*)d_ws;                      // 25.6 MB
  float*    outacc = xproj + (size_t)N_NODES * DIM;     // 25.6 MB
  unsigned* hmax   = (unsigned*)(outacc + (size_t)N_NODES * DIM);
  float*    hsum   = (float*)(hmax + NH);

  gat_init_kernel   <<<2048,         256, 0, stream>>>(outacc, hmax, hsum, N_NODES * DIM);
  gat_gemm_kernel   <<<N_NODES / 16, 256, 0, stream>>>(x, Ww, Wb, xproj, 0);
  gat_logits_kernel <<<N_EDGES / 8,  256, 0, stream>>>(ei, xproj, att, alpha, hmax);
  gat_expsum_kernel <<<1024,         256, 0, stream>>>(alpha, hmax, hsum);
  gat_scatter_kernel<<<N_EDGES / 8,  256, 0, stream>>>(ei, xproj, alpha, hsum, outacc);
  gat_gemm_kernel   <<<N_NODES / 16, 256, 0, stream>>>(outacc, Ow, Ob, out, 1);
}